// SAP_89859305767180
// MI455X (gfx1250) — compile-verified
//
#include <hip/hip_runtime.h>
#include <hip/hip_bf16.h>

#define F_DIM   262144
#define B_DIM   128
#define K_DRAWS 13107.0f
#define EPS     1e-8f

typedef float v2f __attribute__((ext_vector_type(2)));
typedef float v4f __attribute__((ext_vector_type(4)));
typedef float v8f __attribute__((ext_vector_type(8)));

// ---------------------------------------------------------------------------
// Pass 0: zero the 128-float row-sum accumulator in workspace.
// ---------------------------------------------------------------------------
__global__ void sap_zero(float* __restrict__ rowsum) {
    rowsum[threadIdx.x] = 0.0f;
}

// ---------------------------------------------------------------------------
// Pass 1: rowsum[r] = sum_c |x[r][c]| using V_WMMA_F32_16X16X4_F32 with
// B = ones(4x16):  D[m,n] = sum_k A[m,k]*1 + C[m,n].
// A layout (ISA 7.12.2, 32-bit A 16x4): lanes 0-15 hold M=0..15 / K=0,1;
// lanes 16-31 hold M=0..15 / K=2,3 -> each lane loads a contiguous float2.
// Each wave owns 16 rows x 1024 columns (grid 32 x 8 rowgroups, 8 waves/blk).
// ---------------------------------------------------------------------------
__global__ __launch_bounds__(256) void sap_rowsum_wmma(const float* __restrict__ x,
                                                       float* __restrict__ rowsum) {
    const int lane  = threadIdx.x & 31;
    const int wave  = threadIdx.x >> 5;
    const int r0    = blockIdx.y * 16;                 // rowgroup base (8 groups)
    const int gwave = blockIdx.x * 8 + wave;           // 0..255 column partitions
    const int row   = r0 + (lane & 15);
    const int khalf = (lane >> 4) * 2;                 // K offset 0 or 2

    const long cbeg = (long)gwave * (F_DIM / 256);     // 1024 cols per wave
    const long cend = cbeg + (F_DIM / 256);
    const float* base = x + (long)row * F_DIM + khalf;

    v8f acc = {};
    const v2f bones = {1.0f, 1.0f};                    // B = ones (layout-agnostic)

    for (long c = cbeg; c < cend; c += 4) {
        v2f a = *(const v2f*)(base + c);
        __builtin_prefetch(base + c + 64, 0, 3);       // global_prefetch_b8, 256B ahead
        a.x = __builtin_fabsf(a.x);
        a.y = __builtin_fabsf(a.y);
        // 8 args: (neg_a, A, neg_b, B, c_mod, C, reuse_a, reuse_b)
        acc = __builtin_amdgcn_wmma_f32_16x16x4_f32(false, a, false, bones,
                                                    (short)0, acc, false, false);
    }

    // C/D layout: lane L<16 -> VGPR v = (M=v,   N=L);
    //             lane L>=16 -> VGPR v = (M=v+8, N=L-16).  All N identical here.
    if (lane == 0) {
        #pragma unroll
        for (int v = 0; v < 8; ++v) atomicAdd(&rowsum[r0 + v], acc[v]);
    } else if (lane == 16) {
        #pragma unroll
        for (int v = 0; v < 8; ++v) atomicAdd(&rowsum[r0 + 8 + v], acc[v]);
    }
}

// ---------------------------------------------------------------------------
// Pass 2: per element, q = 1-(1-p)^K; keep with prob q (counter-based hash
// RNG keyed on (row,col) -> deterministic); rescale kept values by 1/(q+eps).
// x is L2-resident after pass 1 (134 MB < 192 MB L2); output stored
// non-temporally so the write-once stream does not evict x.
// ---------------------------------------------------------------------------
__device__ __forceinline__ unsigned sap_hash(unsigned v) {
    v ^= v >> 16; v *= 0x7feb352du;
    v ^= v >> 15; v *= 0x846ca68bu;
    v ^= v >> 16;
    return v;
}

__global__ __launch_bounds__(256) void sap_mask_scale(const float* __restrict__ x,
                                                      const float* __restrict__ rowsum,
                                                      float* __restrict__ out) {
    const int row = blockIdx.y;
    const int col = (blockIdx.x * 256 + threadIdx.x) * 4;
    const long idx = (long)row * F_DIM + col;

    v4f xv = *(const v4f*)(x + idx);
    const float invS = 1.0f / (rowsum[row] + EPS);     // uniform scalar load
    const unsigned rseed = (unsigned)row * 0x9E3779B1u ^ 0xC0FFEE42u;

    v4f ov;
    #pragma unroll
    for (int i = 0; i < 4; ++i) {
        float xi = xv[i];
        float p  = __builtin_fabsf(xi) * invS;
        // q = 1 - (1-p)^K, numerically stable for tiny p
        float q  = -expm1f(K_DRAWS * log1pf(-p));
        unsigned h = sap_hash(rseed ^ ((unsigned)(col + i) * 0x85EBCA77u));
        float u  = (float)h * 2.3283064365386963e-10f; // 2^-32
        ov[i] = (u < q) ? xi / (q + EPS) : 0.0f;       // p==0 -> q==0 -> dropped
    }
    __builtin_nontemporal_store(ov, (v4f*)(out + idx));
}

// ---------------------------------------------------------------------------
extern "C" void kernel_launch(void* const* d_in, const int* in_sizes, int n_in,
                              void* d_out, int out_size, void* d_ws, size_t ws_size,
                              hipStream_t stream) {
    (void)in_sizes; (void)n_in; (void)out_size; (void)ws_size;
    const float* x      = (const float*)d_in[0];
    float*       out    = (float*)d_out;
    float*       rowsum = (float*)d_ws;                // 128 floats of scratch

    sap_zero<<<1, B_DIM, 0, stream>>>(rowsum);
    sap_rowsum_wmma<<<dim3(32, B_DIM / 16), 256, 0, stream>>>(x, rowsum);
    sap_mask_scale<<<dim3(F_DIM / 1024, B_DIM), 256, 0, stream>>>(x, rowsum, out);
}